// DetrLoss_72447508349682
// MI455X (gfx1250) — compile-verified
//
#include <hip/hip_runtime.h>
#include <hip/hip_bf16.h>
#include <math.h>
#include <stdint.h>

// Problem constants (from setup_inputs)
constexpr int BB = 32;     // batch
constexpr int QQ = 300;    // queries
constexpr int TT = 64;     // targets
constexpr int CC = 2;      // classes
constexpr int QP = 304;    // queries padded to multiple of 16
constexpr int NT = 256;    // threads per block

constexpr float CLS_SCALE  = 0.1f;
constexpr float BBOX_SCALE = 5.0f;
constexpr float GIOU_SCALE = 2.0f;

typedef __attribute__((ext_vector_type(2))) float v2f;
typedef __attribute__((ext_vector_type(8))) float v8f;

// ---------------------------------------------------------------------------
// Kernel 0: zero the global accumulators
// ---------------------------------------------------------------------------
__global__ void init_kernel(double* acc) {
    if (threadIdx.x < 4) acc[threadIdx.x] = 0.0;
}

// ---------------------------------------------------------------------------
// Kernel 1: per-batch cost matrix, cost[b][q][t] (QP x T padded, row major)
//   phase 1: cost_bbox - giou  -> LDS (pad rows = 0)
//   phase 2: += cost_class via V_WMMA_F32_16X16X4_F32 fused accumulate,
//            C operand from LDS, D written once to global (padded, no branches)
//   cost_class = (-probs[QxC, C padded to 4]) x onehot(labels)[4xT]
// ---------------------------------------------------------------------------
__global__ void cost_kernel(const float* __restrict__ logits,
                            const float* __restrict__ pred,
                            const float* __restrict__ tbox,
                            const int*   __restrict__ tlabIn,
                            float*       __restrict__ cost) {
    const int b = blockIdx.x, tid = threadIdx.x;

    extern __shared__ float scost[];   // QP*TT floats (76 KB dynamic LDS)

    __shared__ float pcx[QP], pcy[QP], pw[QP], ph[QP];
    __shared__ float pr0[QP], pr1[QP];
    __shared__ float tcx[TT], tcy[TT], tw[TT], th[TT];
    __shared__ int   tlab[TT];

    // load targets
    for (int t = tid; t < TT; t += NT) {
        const float* tb = tbox + ((size_t)b * TT + t) * 4;
        tcx[t] = tb[0]; tcy[t] = tb[1]; tw[t] = tb[2]; th[t] = tb[3];
        tlab[t] = tlabIn[(size_t)b * TT + t];
    }
    // load preds + softmax probs (C=2, stable)
    for (int q = tid; q < QP; q += NT) {
        if (q < QQ) {
            const float* pb = pred + ((size_t)b * QQ + q) * 4;
            pcx[q] = pb[0]; pcy[q] = pb[1]; pw[q] = pb[2]; ph[q] = pb[3];
            float l0 = logits[((size_t)b * QQ + q) * CC + 0];
            float l1 = logits[((size_t)b * QQ + q) * CC + 1];
            float mx = fmaxf(l0, l1);
            float e0 = expf(l0 - mx), e1 = expf(l1 - mx);
            float s  = e0 + e1;
            pr0[q] = e0 / s; pr1[q] = e1 / s;
        } else {
            pcx[q] = pcy[q] = pw[q] = ph[q] = 0.f;
            pr0[q] = pr1[q] = 0.f;
        }
    }
    __syncthreads();

    // phase 1: L1 - GIoU into LDS (pad rows set to 0, keeps NaN out of WMMA C)
    for (int idx = tid; idx < QP * TT; idx += NT) {
        int q = idx >> 6, t = idx & 63;
        float val = 0.0f;
        if (q < QQ) {
            float al = pcx[q] - 0.5f * pw[q], at = pcy[q] - 0.5f * ph[q];
            float ar = pcx[q] + 0.5f * pw[q], ab = pcy[q] + 0.5f * ph[q];
            float bl = tcx[t] - 0.5f * tw[t], bt = tcy[t] - 0.5f * th[t];
            float br = tcx[t] + 0.5f * tw[t], bbm = tcy[t] + 0.5f * th[t];
            float area1 = (ar - al) * (ab - at);
            float area2 = (br - bl) * (bbm - bt);
            float iw = fmaxf(fminf(ar, br) - fmaxf(al, bl), 0.f);
            float ih = fmaxf(fminf(ab, bbm) - fmaxf(at, bt), 0.f);
            float inter = iw * ih;
            float uni = area1 + area2 - inter;
            float iou = inter / uni;
            float ew = fmaxf(fmaxf(ar, br) - fminf(al, bl), 0.f);
            float eh = fmaxf(fmaxf(ab, bbm) - fminf(at, bt), 0.f);
            float enc = ew * eh;
            float giou = iou - (enc - uni) / enc;
            float cbb = fabsf(pcx[q] - tcx[t]) + fabsf(pcy[q] - tcy[t]) +
                        fabsf(pw[q]  - tw[t])  + fabsf(ph[q]  - th[t]);
            val = cbb - giou;   // COST_BBOX*l1 + COST_GIOU*(-giou)
        }
        scost[idx] = val;
    }
    __syncthreads();

    // phase 2: fused class-cost accumulate via WMMA f32 16x16x4
    float* cbp = cost + (size_t)b * QP * TT;
    const int wave = tid >> 5, lane = tid & 31;
    const int tilesT = TT / 16;              // 4
    const int nTiles = (QP / 16) * tilesT;   // 19*4 = 76
    const int n    = lane & 15;
    const int mofs = (lane >= 16) ? 8 : 0;

    for (int tile = wave; tile < nTiles; tile += NT / 32) {
        int tq = tile / tilesT, tt = tile % tilesT;

        // A: 16x4 f32. lanes 0-15 -> M=lane, K=0,1 ; lanes 16-31 -> K=2,3 (zero pad)
        v2f a; v2f bm;
        if (lane < 16) {
            int q = tq * 16 + lane;
            a.x = -pr0[q]; a.y = -pr1[q];
            int t = tt * 16 + lane;
            int L = tlab[t];
            bm.x = (L == 0) ? 1.0f : 0.0f;   // B row K=0 = onehot(label==0)
            bm.y = (L == 1) ? 1.0f : 0.0f;   // B row K=1 = onehot(label==1)
        } else {
            a.x = 0.f; a.y = 0.f;            // K=2,3 padding
            bm.x = 0.f; bm.y = 0.f;
        }

        // C: 16x16 f32, VGPR r: lanes 0-15 -> (M=r, N=lane); lanes 16-31 -> (M=r+8, N=lane-16)
        v8f c;
        for (int r = 0; r < 8; ++r) {
            int q = tq * 16 + r + mofs;
            int t = tt * 16 + n;
            c[r] = scost[q * TT + t];        // always in bounds (QP padded)
        }

        v8f d = __builtin_amdgcn_wmma_f32_16x16x4_f32(
            /*neg_a=*/false, a, /*neg_b=*/false, bm,
            /*c_mod=*/(short)0, c, /*reuse_a=*/false, /*reuse_b=*/false);

        for (int r = 0; r < 8; ++r) {
            int q = tq * 16 + r + mofs;
            int t = tt * 16 + n;
            cbp[q * TT + t] = d[r];          // unconditional store (padded buffer)
        }
    }
}

// ---------------------------------------------------------------------------
// Kernel 2: Hungarian (Jonker-Volgenant) per batch, float64, block-parallel.
// Solves cost_block.T (n=T rows, m=Q cols) exactly as reference _lsa.
// cost_T[i][j] = cost[b][j][i]   (i: target row, j: query col)
// ---------------------------------------------------------------------------
__global__ void lsa_kernel(const float* __restrict__ cost, int* __restrict__ src) {
    const int b = blockIdx.x, tid = threadIdx.x;
    const int n = TT, m = QQ;
    const float* cb = cost + (size_t)b * QP * TT;   // padded stride
    const double INF = 1e300;

    __shared__ double u[TT + 1];
    __shared__ double v[QQ + 1];
    __shared__ double minv[QQ + 1];
    __shared__ int    p[QQ + 1];
    __shared__ int    way[QQ + 1];
    __shared__ unsigned char used[QQ + 1];
    __shared__ double red_val[NT];
    __shared__ int    red_idx[NT];
    __shared__ int    s_j1;

    for (int j = tid; j <= m; j += NT) { v[j] = 0.0; p[j] = 0; way[j] = 0; }
    for (int j = tid; j <= n; j += NT) u[j] = 0.0;
    __syncthreads();

    for (int i = 1; i <= n; ++i) {
        if (tid == 0) p[0] = i;
        for (int j = tid; j <= m; j += NT) { minv[j] = INF; used[j] = 0; }
        __syncthreads();

        int j0 = 0;
        while (true) {
            used[j0] = 1;  // all threads write same value (benign)
            __syncthreads();
            int i0 = p[j0];
            double ui0 = u[i0];

            // update minv over free columns
            for (int j = 1 + tid; j <= m; j += NT) {
                if (!used[j]) {
                    double cur = (double)cb[(size_t)(j - 1) * TT + (i0 - 1)] - ui0 - v[j];
                    if (cur < minv[j]) { minv[j] = cur; way[j] = j0; }
                }
            }
            __syncthreads();

            // argmin over free columns, first-occurrence tie-break (np.argmin)
            double bv = INF; int bj = m;
            for (int j = 1 + tid; j <= m; j += NT) {
                double val = used[j] ? INF : minv[j];
                if (val < bv || (val == bv && j < bj)) { bv = val; bj = j; }
            }
            red_val[tid] = bv; red_idx[tid] = bj;
            __syncthreads();
            for (int s = NT / 2; s > 0; s >>= 1) {
                if (tid < s) {
                    double ov = red_val[tid + s]; int oj = red_idx[tid + s];
                    if (ov < red_val[tid] ||
                        (ov == red_val[tid] && oj < red_idx[tid])) {
                        red_val[tid] = ov; red_idx[tid] = oj;
                    }
                }
                __syncthreads();
            }
            int j1 = red_idx[0];
            double delta = red_val[0];
            if (tid == 0) s_j1 = j1;

            // dual updates (p[j] distinct over used j -> race-free)
            for (int j = tid; j <= m; j += NT) {
                if (used[j]) { u[p[j]] += delta; v[j] -= delta; }
                else           minv[j] -= delta;
            }
            __syncthreads();
            j0 = s_j1;
            if (p[j0] == 0) break;   // uniform
        }

        // traceback (serial)
        if (tid == 0) {
            int jj = j0;
            while (jj) { int jn = way[jj]; p[jj] = p[jn]; jj = jn; }
        }
        __syncthreads();
    }

    for (int j = 1 + tid; j <= m; j += NT) {
        int r = p[j];
        if (r > 0) src[(size_t)b * TT + (r - 1)] = j - 1;
    }
}

// ---------------------------------------------------------------------------
// Kernel 3: losses -> 4 double accumulators {sum(wt*nll), sum(wt), sum l1, sum giou}
// ---------------------------------------------------------------------------
__global__ void loss_kernel(const float* __restrict__ logits,
                            const float* __restrict__ pred,
                            const float* __restrict__ tbox,
                            const int*   __restrict__ labels,
                            const int*   __restrict__ src,
                            double*      __restrict__ acc) {
    const int b = blockIdx.x, tid = threadIdx.x;
    __shared__ int tc[QQ];
    __shared__ double red[NT];

    for (int q = tid; q < QQ; q += NT) tc[q] = CC - 1;   // "no object"
    __syncthreads();
    for (int t = tid; t < TT; t += NT) {
        int q = src[(size_t)b * TT + t];
        tc[q] = labels[(size_t)b * TT + t];              // matches are unique
    }
    __syncthreads();

    double a_wnll = 0.0, a_w = 0.0, a_l1 = 0.0, a_g = 0.0;

    // weighted CE over all queries
    for (int q = tid; q < QQ; q += NT) {
        float l0 = logits[((size_t)b * QQ + q) * CC + 0];
        float l1 = logits[((size_t)b * QQ + q) * CC + 1];
        float mx = fmaxf(l0, l1);
        float lse = mx + logf(expf(l0 - mx) + expf(l1 - mx));
        int c = tc[q];
        float lc = (c == 0) ? l0 : l1;
        float nll = lse - lc;
        float wt = (c == CC - 1) ? CLS_SCALE : 1.0f;
        a_wnll += (double)(wt * nll);
        a_w    += (double)wt;
    }

    // matched box losses
    for (int t = tid; t < TT; t += NT) {
        int q = src[(size_t)b * TT + t];
        const float* pb = pred + ((size_t)b * QQ + q) * 4;
        const float* tb = tbox + ((size_t)b * TT + t) * 4;
        float l1 = fabsf(pb[0] - tb[0]) + fabsf(pb[1] - tb[1]) +
                   fabsf(pb[2] - tb[2]) + fabsf(pb[3] - tb[3]);
        float al = pb[0] - 0.5f * pb[2], at = pb[1] - 0.5f * pb[3];
        float ar = pb[0] + 0.5f * pb[2], ab = pb[1] + 0.5f * pb[3];
        float bl = tb[0] - 0.5f * tb[2], bt = tb[1] - 0.5f * tb[3];
        float br = tb[0] + 0.5f * tb[2], bbm = tb[1] + 0.5f * tb[3];
        float area1 = (ar - al) * (ab - at);
        float area2 = (br - bl) * (bbm - bt);
        float iw = fmaxf(fminf(ar, br) - fmaxf(al, bl), 0.f);
        float ih = fmaxf(fminf(ab, bbm) - fmaxf(at, bt), 0.f);
        float inter = iw * ih;
        float uni = area1 + area2 - inter;
        float iou = inter / uni;
        float ew = fmaxf(fmaxf(ar, br) - fminf(al, bl), 0.f);
        float eh = fmaxf(fmaxf(ab, bbm) - fminf(at, bt), 0.f);
        float enc = ew * eh;
        float giou = iou - (enc - uni) / enc;
        a_l1 += (double)l1;
        a_g  += (double)giou;
    }

    // block reduce each accumulator, atomicAdd to global
    double vals[4] = {a_wnll, a_w, a_l1, a_g};
    for (int k = 0; k < 4; ++k) {
        red[tid] = vals[k];
        __syncthreads();
        for (int s = NT / 2; s > 0; s >>= 1) {
            if (tid < s) red[tid] += red[tid + s];
            __syncthreads();
        }
        if (tid == 0) atomicAdd(&acc[k], red[0]);
        __syncthreads();
    }
}

// ---------------------------------------------------------------------------
// Kernel 4: finalize scalar loss
// ---------------------------------------------------------------------------
__global__ void finalize_kernel(const double* __restrict__ acc, float* __restrict__ out) {
    if (threadIdx.x == 0) {
        const double nb = (double)BB * (double)TT;     // num_bboxes = 2048
        double ce = acc[0] / acc[1];
        double l1 = acc[2] / nb;
        double gl = (nb - acc[3]) / nb;                // mean(1 - giou)
        out[0] = (float)(ce + (double)BBOX_SCALE * l1 + (double)GIOU_SCALE * gl);
    }
}

// ---------------------------------------------------------------------------
extern "C" void kernel_launch(void* const* d_in, const int* in_sizes, int n_in,
                              void* d_out, int out_size, void* d_ws, size_t ws_size,
                              hipStream_t stream) {
    (void)in_sizes; (void)n_in; (void)out_size; (void)ws_size;

    const float* logits = (const float*)d_in[0];  // (B,Q,C)
    const float* pred   = (const float*)d_in[1];  // (B,Q,4)
    const float* tbox   = (const float*)d_in[2];  // (B,T,4)
    const int*   labels = (const int*)d_in[3];    // (B,T)

    float*  d_cost = (float*)d_ws;                                  // B*QP*T f32 (padded)
    int*    d_src  = (int*)((char*)d_ws + sizeof(float) * BB * QP * TT);
    double* d_acc  = (double*)((char*)d_src + sizeof(int) * BB * TT);

    const size_t lds_bytes = sizeof(float) * QP * TT;   // 76 KB dynamic LDS

    init_kernel<<<1, 32, 0, stream>>>(d_acc);
    cost_kernel<<<BB, NT, lds_bytes, stream>>>(logits, pred, tbox, labels, d_cost);
    lsa_kernel<<<BB, NT, 0, stream>>>(d_cost, d_src);
    loss_kernel<<<BB, NT, 0, stream>>>(logits, pred, tbox, labels, d_src, d_acc);
    finalize_kernel<<<1, 32, 0, stream>>>(d_acc, (float*)d_out);
}